// FocalLoss_15968688407054
// MI455X (gfx1250) — compile-verified
//
#include <hip/hip_runtime.h>

// Focal BCE loss, MI455X (gfx1250, wave32).
// Memory-bound: 320 MB read-once -> ~13.7us floor at 23.3 TB/s.
// Weighted attribute reduction runs on the matrix unit:
// V_WMMA_F32_16X16X4_F32 with B = broadcast(-ln2 * attr weights), C
// accumulated across the whole batch loop (no cross-lane traffic until
// the epilogue). Per-element math trimmed to raw v_log_f32 (log2) +
// packed f32 ops so VALU stays well under the HBM roofline.

typedef float v2f __attribute__((ext_vector_type(2)));
typedef float v8f __attribute__((ext_vector_type(8)));

#define EPS_F 1e-12f
#define NEG_LN2 (-0.69314718055994530942f)

// Returns om^2 * log2(x) for a pair; the -ln2 (and per-attr weight) are
// folded into the WMMA B-matrix. x is always in [1e-12, 1]: normal, finite,
// so the raw hardware log2 needs no denorm/inf fixup.
__device__ __forceinline__ v2f focal_pair(v2f p, int t0, int t1) {
    v2f om;                           // 1-p (unclipped; reference focal term)
    om.x = 1.0f - p.x;
    om.y = 1.0f - p.y;
    // t==1: log(max(p,eps));  t==0: log(1-p)  (f32 upper clip is a no-op)
    float x0 = t0 ? fmaxf(p.x, EPS_F) : om.x;
    float x1 = t1 ? fmaxf(p.y, EPS_F) : om.y;
    float l0 = __builtin_amdgcn_logf(x0);   // raw v_log_f32 (log2)
    float l1 = __builtin_amdgcn_logf(x1);
    v2f f;
    f.x = (om.x * om.x) * l0;
    f.y = (om.y * om.y) * l1;
    return f;
}

// inputs:  [B, 40] f32 (160B rows, 16B aligned)
// targets: [40, B] i32 (contiguous in sample index per attr -> coalesced)
// A hardcoded to 40 (10 K-blocks of 4) to match the reference.
__global__ __launch_bounds__(256) void focal_wmma_partial(
    const float* __restrict__ inputs,
    const int*   __restrict__ targets,
    const float* __restrict__ w,
    float*       __restrict__ partials,
    int B)
{
    const int lane = threadIdx.x & 31;
    const int wave = threadIdx.x >> 5;
    const int wavesPerBlock = blockDim.x >> 5;
    const int gwave  = blockIdx.x * wavesPerBlock + wave;
    const int nwaves = gridDim.x * wavesPerBlock;
    const int m      = lane & 15;   // sample row within 16-row tile
    const int half   = lane >> 4;   // lanes 0-15 -> k{0,1}, lanes 16-31 -> k{2,3}
    const int ntiles = B >> 4;      // B is a multiple of 16

    // B-matrix weight pairs: B[k,n] = -ln2 * w[4*kb+k] for every column n.
    v2f wp[10];
#pragma unroll
    for (int kb = 0; kb < 10; ++kb) {
        const int a0 = kb * 4 + 2 * half;
        wp[kb].x = w[a0]     * NEG_LN2;
        wp[kb].y = w[a0 + 1] * NEG_LN2;
    }

    // Wave-uniform (SGPR) target base pointers per K-block: targets + 4*kb*B.
    // Hoisted out of the tile loop; per-tile address math is two 32-bit adds.
    const int* tb[10];
#pragma unroll
    for (int kb = 0; kb < 10; ++kb)
        tb[kb] = targets + (size_t)(4 * kb) * (size_t)B;

    const unsigned voff0 = (unsigned)(2 * half) * (unsigned)B; // lane-dep attr part
    const unsigned voff1 = voff0 + (unsigned)B;

    v8f c = {0.f, 0.f, 0.f, 0.f, 0.f, 0.f, 0.f, 0.f};

    // Wave-uniform grid-stride over 16-sample tiles (EXEC stays all-1s for WMMA).
    for (int tile = gwave; tile < ntiles; tile += nwaves) {
        const unsigned i  = (unsigned)tile * 16u + (unsigned)m;  // sample index
        const unsigned o0 = voff0 + i;
        const unsigned o1 = voff1 + i;
        const float* row = inputs + (size_t)i * 40;              // 16B-aligned row
#pragma unroll
        for (int kb = 0; kb < 10; ++kb) {
            const int a0 = kb * 4 + 2 * half;
            v2f p  = __builtin_nontemporal_load((const v2f*)(row + a0));
            int t0 = __builtin_nontemporal_load(tb[kb] + o0);    // coalesced
            int t1 = __builtin_nontemporal_load(tb[kb] + o1);
            v2f f  = focal_pair(p, t0, t1);
            // D = A(16x4 F-tile) x B(4x16 weight-broadcast) + C; C carried
            // across kb and all tiles: the matrix unit owns the reduction.
            c = __builtin_amdgcn_wmma_f32_16x16x4_f32(
                    false, f, false, wp[kb], (short)0, c, false, false);
        }
    }

    // Epilogue: sum all 16x16 C elements of this wave (= 16x wave total).
    float s = c[0] + c[1] + c[2] + c[3] + c[4] + c[5] + c[6] + c[7];
#pragma unroll
    for (int off = 16; off >= 1; off >>= 1)
        s += __shfl_xor(s, off, 32);

    __shared__ float smem[32];
    if (lane == 0) smem[wave] = s;
    __syncthreads();
    if (wave == 0) {
        float bs = (lane < wavesPerBlock) ? smem[lane] : 0.f;
#pragma unroll
        for (int off = 16; off >= 1; off >>= 1)
            bs += __shfl_xor(bs, off, 32);
        if (lane == 0) partials[blockIdx.x] = bs;
    }
}

// Deterministic fixed-order final reduction (no float atomics).
__global__ __launch_bounds__(512) void reduce_partials_kernel(
    const float* __restrict__ partials, int n, float* __restrict__ out, float scale)
{
    const int lane = threadIdx.x & 31;
    const int wave = threadIdx.x >> 5;
    float s = 0.f;
    for (int idx = threadIdx.x; idx < n; idx += 512) s += partials[idx];
#pragma unroll
    for (int off = 16; off >= 1; off >>= 1) s += __shfl_xor(s, off, 32);

    __shared__ float smem[16];
    if (lane == 0) smem[wave] = s;
    __syncthreads();
    if (wave == 0) {
        float t = (lane < 16) ? smem[lane] : 0.f;
#pragma unroll
        for (int off = 16; off >= 1; off >>= 1) t += __shfl_xor(t, off, 32);
        if (lane == 0) out[0] = t * scale;
    }
}

extern "C" void kernel_launch(void* const* d_in, const int* in_sizes, int n_in,
                              void* d_out, int out_size, void* d_ws, size_t ws_size,
                              hipStream_t stream) {
    const float* inputs  = (const float*)d_in[0];   // [B, A] f32
    const int*   targets = (const int*)  d_in[1];   // [A, B] i32
    const float* w       = (const float*)d_in[2];   // [A]    f32

    const int A = in_sizes[2];            // 40
    const int B = in_sizes[0] / A;        // 1,000,000

    float* partials = (float*)d_ws;       // 1000 floats of scratch (4 KB)
    const int blocks = 1000;              // 8000 waves -> ~8 tiles per wave

    focal_wmma_partial<<<blocks, 256, 0, stream>>>(inputs, targets, w, partials, B);

    // scale: /16 (WMMA column duplication) and /A (per-sample mean over attrs)
    reduce_partials_kernel<<<1, 512, 0, stream>>>(
        partials, blocks, (float*)d_out, 1.0f / (16.0f * (float)A));
}